// SetCriterionBAST_23785528885873
// MI455X (gfx1250) — compile-verified
//
#include <hip/hip_runtime.h>
#include <math.h>

// Problem constants (match reference)
#define BB 128
#define KK 2048
#define NN 4
#define CC 64
#define DD 3

typedef float v2f __attribute__((ext_vector_type(2)));
typedef float v8f __attribute__((ext_vector_type(8)));

__device__ __forceinline__ float focal_pos_f(float x) {
    // ALPHA * softplus(-x) * (1 - sigmoid(x))^2, numerically stable
    float sp = fmaxf(-x, 0.0f) + log1pf(expf(-fabsf(x)));
    float t  = 1.0f / (1.0f + expf(x));   // = 1 - sigmoid(x)
    return 0.25f * sp * t * t;
}

__device__ __forceinline__ float focal_bce_f(float x, float tgt) {
    float p  = 1.0f / (1.0f + expf(-x));
    float ce = fmaxf(x, 0.0f) - x * tgt + log1pf(expf(-fabsf(x)));
    float pt = p * tgt + (1.0f - p) * (1.0f - tgt);
    float at = 0.25f * tgt + 0.75f * (1.0f - tgt);
    float om = 1.0f - pt;
    return at * ce * om * om;
}

__device__ __forceinline__ void top4_insert(float c[4], int idx[4], float v, int k) {
    if (v >= c[3]) return;
    c[3] = v; idx[3] = k;
#pragma unroll
    for (int j = 3; j > 0; --j) {
        if (c[j] < c[j - 1]) {
            float tc = c[j]; c[j] = c[j - 1]; c[j - 1] = tc;
            int   ti = idx[j]; idx[j] = idx[j - 1]; idx[j - 1] = ti;
        }
    }
}

// One block per batch: WMMA cls-cost, fused loc/obj cost, top-4 tracking,
// exact tiny-Hungarian, matched losses -> per-batch partial sums.
__global__ __launch_bounds__(512)
void set_criterion_batch_kernel(const float* __restrict__ loc_out,
                                const float* __restrict__ obj_logit,
                                const float* __restrict__ cls_logit,
                                const float* __restrict__ gt_loc,
                                const float* __restrict__ gt_cls,
                                float* __restrict__ partials) {
    const int b    = blockIdx.x;
    const int tid  = threadIdx.x;          // 0..511 (16 waves of 32)
    const int wave = tid >> 5;
    const int lane = tid & 31;
    const int m    = lane & 15;            // WMMA M / N index within tile
    const int h    = lane >> 4;            // half-wave selector

    const float* clsb = cls_logit + (size_t)b * KK * CC;
    const float* locb = loc_out   + (size_t)b * KK * DD;
    const float* objb = obj_logit + (size_t)b * KK;

    // s_gt padded to 16 rows so the WMMA B operand is an unconditional
    // 8B-aligned ds_load_b64 for every lane (rows 4..15 are zero).
    __shared__ float s_gt[16 * CC];        // rows 0..3 = gt_cls[b], 4..15 = 0
    __shared__ float s_vt[NN][4];          // normalized gt_loc[b]
    __shared__ float s_candc[NN][128];     // per-n candidate costs
    __shared__ int   s_candi[NN][128];     // per-n candidate k indices
    __shared__ float s_red[512];
    __shared__ float s_small[8];
    __shared__ int   s_pred[NN];

    // ---- stage GT data ----
    if (tid < NN * CC) s_gt[tid] = gt_cls[(size_t)b * NN * CC + tid];
    for (int j = tid + NN * CC; j < 16 * CC; j += 512) s_gt[j] = 0.0f;
    if (tid < NN) {
        float gx = gt_loc[((size_t)b * NN + tid) * DD + 0];
        float gy = gt_loc[((size_t)b * NN + tid) * DD + 1];
        float gz = gt_loc[((size_t)b * NN + tid) * DD + 2];
        float nr = fmaxf(sqrtf(gx * gx + gy * gy + gz * gz), 1e-8f);
        float iv = 1.0f / nr;
        s_vt[tid][0] = gx * iv; s_vt[tid][1] = gy * iv; s_vt[tid][2] = gz * iv;
    }
    __syncthreads();

    float vtx = 0.f, vty = 0.f, vtz = 0.f;
    if (m < NN) { vtx = s_vt[m][0]; vty = s_vt[m][1]; vtz = s_vt[m][2]; }

    float c4[4] = {3.0e38f, 3.0e38f, 3.0e38f, 3.0e38f};
    int   i4[4] = {-1, -1, -1, -1};
    float objacc = 0.0f;

    // ---- stream K: 8 tiles of 16 rows per wave ----
    for (int tt = 0; tt < 8; ++tt) {
        const int k0 = (wave * 8 + tt) * 16;
        // lane(m,h) owns channels {4s+2h, 4s+2h+1} of cls row k0+m (A) and
        // of gt row m (B); both are 8-byte aligned float2 streams.
        const v2f* arow = (const v2f*)(clsb + (size_t)(k0 + m) * CC + 2 * h);
        const v2f* brow = (const v2f*)(&s_gt[m * CC + 2 * h]);

        v8f acc = {};
#pragma unroll
        for (int s = 0; s < 16; ++s) {
            v2f xv = arow[2 * s];
            v2f a; a.x = focal_pos_f(xv.x); a.y = focal_pos_f(xv.y);
            v2f bv = brow[2 * s];
            acc = __builtin_amdgcn_wmma_f32_16x16x4_f32(
                false, a, false, bv, (short)0, acc, false, false);
        }

        // D layout: lane(m,h) VGPR r -> dot[k0 + r + 8h, n=m]
        if (m < NN) {
#pragma unroll
            for (int r = 0; r < 8; ++r) {
                int k = k0 + 8 * h + r;
                float lx = locb[k * 3 + 0];
                float ly = locb[k * 3 + 1];
                float lz = locb[k * 3 + 2];
                float nr = fmaxf(sqrtf(lx * lx + ly * ly + lz * lz), 1e-8f);
                float iv = 1.0f / nr;
                float cv = (lx * vtx + ly * vty + lz * vtz) * iv;
                cv = fminf(fmaxf(cv, -1.0f + 1e-6f), 1.0f - 1e-6f);
                float lc  = acosf(cv) * 0.3183098861837907f;   // /pi
                float o   = objb[k];
                float sig = 1.0f / (1.0f + expf(-o));
                float cost = lc + 0.25f * acc[r] - 0.1f * sig;  // LOC/CLS/OBJ cost weights
                top4_insert(c4, i4, cost, k);
                if (m == 0)   // lanes (m=0,h=0/1) cover every k exactly once
                    objacc += fmaxf(o, 0.0f) + log1pf(expf(-fabsf(o)));
            }
        }
    }

    // ---- publish per-lane candidates & obj partial ----
    if (m < NN) {
        int slot = (wave * 2 + h) * 4;
#pragma unroll
        for (int e = 0; e < 4; ++e) { s_candc[m][slot + e] = c4[e]; s_candi[m][slot + e] = i4[e]; }
    }
    s_red[tid] = objacc;
    __syncthreads();
    for (int off = 256; off > 0; off >>= 1) {
        if (tid < off) s_red[tid] += s_red[tid + off];
        __syncthreads();
    }

    // ---- final top-4 per n ----
    if (tid < NN) {
        float fc[4] = {3.0e38f, 3.0e38f, 3.0e38f, 3.0e38f};
        int   fi[4] = {-1, -1, -1, -1};
        for (int j = 0; j < 128; ++j) top4_insert(fc, fi, s_candc[tid][j], s_candi[tid][j]);
#pragma unroll
        for (int e = 0; e < 4; ++e) { s_candc[tid][e] = fc[e]; s_candi[tid][e] = fi[e]; }
    }
    __syncthreads();

    // ---- exact assignment: each row restricted to its top-4 (provably optimal) ----
    if (tid == 0) {
        float bc = 3.0e38f; int b0 = 0, b1 = 0, b2 = 0, b3 = 0;
        for (int q0 = 0; q0 < 4; ++q0) {
            int a0 = s_candi[0][q0]; float t0 = s_candc[0][q0];
            for (int q1 = 0; q1 < 4; ++q1) {
                int a1 = s_candi[1][q1]; if (a1 == a0) continue;
                float t1 = t0 + s_candc[1][q1];
                for (int q2 = 0; q2 < 4; ++q2) {
                    int a2 = s_candi[2][q2]; if (a2 == a0 || a2 == a1) continue;
                    float t2 = t1 + s_candc[2][q2];
                    for (int q3 = 0; q3 < 4; ++q3) {
                        int a3 = s_candi[3][q3]; if (a3 == a0 || a3 == a1 || a3 == a2) continue;
                        float t3 = t2 + s_candc[3][q3];
                        if (t3 < bc) { bc = t3; b0 = a0; b1 = a1; b2 = a2; b3 = a3; }
                    }
                }
            }
        }
        s_pred[0] = b0; s_pred[1] = b1; s_pred[2] = b2; s_pred[3] = b3;
        // obj BCE: sum ce(target=0) over all k, then ce1-ce0 = -x on matched
        float adj = objb[b0] + objb[b1] + objb[b2] + objb[b3];
        partials[b * 3 + 2] = s_red[0] - adj;
    }
    __syncthreads();

    // ---- matched losses ----
    float clsv = 0.0f;
    if (tid < NN * CC) {
        int n = tid >> 6, c = tid & 63;
        int pk = s_pred[n];
        float x   = clsb[(size_t)pk * CC + c];
        float tgt = s_gt[tid];
        clsv = focal_bce_f(x, tgt);
    }
    s_red[tid] = clsv;
    if (tid < NN) {
        int pk = s_pred[tid];
        float acc_l = 0.0f;
#pragma unroll
        for (int d = 0; d < DD; ++d) {
            float diff = locb[pk * 3 + d] - gt_loc[((size_t)b * NN + tid) * DD + d];
            acc_l += diff * diff;
        }
        s_small[tid] = acc_l;
    }
    __syncthreads();
    for (int off = 256; off > 0; off >>= 1) {
        if (tid < off) s_red[tid] += s_red[tid + off];
        __syncthreads();
    }
    if (tid == 0) {
        partials[b * 3 + 0] = s_small[0] + s_small[1] + s_small[2] + s_small[3];
        partials[b * 3 + 1] = s_red[0];
    }
}

__global__ __launch_bounds__(128)
void set_criterion_reduce_kernel(const float* __restrict__ partials,
                                 float* __restrict__ out) {
    __shared__ float sl[128], sc[128], so[128];
    int t = threadIdx.x;
    sl[t] = partials[t * 3 + 0];
    sc[t] = partials[t * 3 + 1];
    so[t] = partials[t * 3 + 2];
    __syncthreads();
    for (int off = 64; off > 0; off >>= 1) {
        if (t < off) { sl[t] += sl[t + off]; sc[t] += sc[t + off]; so[t] += so[t + off]; }
        __syncthreads();
    }
    if (t == 0) {
        // means: loc /(B*N*D)=1536, cls /(B*N*C)=32768, obj /(B*K)=262144
        out[0] = sl[0] * (1.0f / 1536.0f)
               + sc[0] * (1.0f / 32768.0f)
               + so[0] * (1.0f / 262144.0f);
    }
}

extern "C" void kernel_launch(void* const* d_in, const int* in_sizes, int n_in,
                              void* d_out, int out_size, void* d_ws, size_t ws_size,
                              hipStream_t stream) {
    (void)in_sizes; (void)n_in; (void)out_size; (void)ws_size;
    const float* loc_out   = (const float*)d_in[0];
    const float* obj_logit = (const float*)d_in[1];
    const float* cls_logit = (const float*)d_in[2];
    const float* gt_loc    = (const float*)d_in[3];
    const float* gt_cls    = (const float*)d_in[4];
    float* out      = (float*)d_out;
    float* partials = (float*)d_ws;   // BB*3 floats

    set_criterion_batch_kernel<<<dim3(BB), dim3(512), 0, stream>>>(
        loc_out, obj_logit, cls_logit, gt_loc, gt_cls, partials);
    set_criterion_reduce_kernel<<<dim3(1), dim3(128), 0, stream>>>(partials, out);
}